// LearnableSOn_84258668413402
// MI455X (gfx1250) — compile-verified
//
#include <hip/hip_runtime.h>

// ---------------------------------------------------------------------------
// expm(A - A^T) for N=2048 fp32 via scaling-and-squaring Taylor (Horner),
// with all GEMMs running on the CDNA5 fp32 matrix pipe (V_WMMA_F32_16X16X4_F32).
//
//   X  = (A - A^T) * 2^-4            (||X|| ~ 0.08)
//   P  = I + X/8                     (Horner seed, k=8)
//   for k = 7..1:  P <- I + (1/k) * X * P      (7 WMMA GEMMs)
//   square 4x:     P <- P * P                  (4 WMMA GEMMs)
//
// ~11 GEMMs x 17.2 GFLOP; all three 16.8MB matrices are L2-resident (192MB),
// so the problem is pure matrix-core throughput -> fp32 WMMA is the right tool
// (bf16/f16 WMMA would wreck fp32 validation accuracy over 11 chained GEMMs).
// ---------------------------------------------------------------------------

#define NMAT 2048
#define SQUARINGS 4          // scale by 2^-4
#define HORNER_TOP 8         // Taylor order 8

typedef float v2f __attribute__((ext_vector_type(2)));
typedef float v8f __attribute__((ext_vector_type(8)));

// ---------------------------------------------------------------------------
// Elementwise: X = (A - A^T) * invScale ;  P = I + X * (1/HORNER_TOP)
// ---------------------------------------------------------------------------
__global__ __launch_bounds__(256)
void skew_init_kernel(const float* __restrict__ A,
                      float* __restrict__ X,
                      float* __restrict__ P)
{
    const int idx = blockIdx.x * 256 + threadIdx.x;        // N*N/256 blocks
    const int i = idx >> 11;            // row  (N = 2048 = 1<<11)
    const int j = idx & (NMAT - 1);     // col
    const float invScale = 1.0f / (float)(1 << SQUARINGS);
    float s = (A[i * NMAT + j] - A[j * NMAT + i]) * invScale;
    X[idx] = s;
    P[idx] = s * (1.0f / (float)HORNER_TOP) + ((i == j) ? 1.0f : 0.0f);
}

// ---------------------------------------------------------------------------
// D = alpha * (Amat x Bmat) [+ I]
// 128x128 C-block per 256-thread workgroup (8 waves).
// Each wave: 32 rows x 64 cols = 2x4 tiles of 16x16, K stepped 4 at a time
// through V_WMMA_F32_16X16X4_F32.
// ---------------------------------------------------------------------------
__global__ __launch_bounds__(256)
void gemm_wmma_f32_kernel(const float* __restrict__ Amat,
                          const float* __restrict__ Bmat,
                          float* __restrict__ Dmat,
                          float alpha, int addIdentity)
{
    // LDS tiles: stride 36 floats = 144B (16B aligned, conflict-skewed).
    __shared__ float As[128][36];   // [row][k]
    __shared__ float Bs[128][36];   // [col][k]  (transposed on load)

    const int tid  = threadIdx.x;
    const int wave = tid >> 5;
    const int lane = tid & 31;
    const int half = lane >> 4;       // 0: lanes 0-15, 1: lanes 16-31
    const int r    = lane & 15;

    const int rowBase = blockIdx.y * 128;
    const int colBase = blockIdx.x * 128;
    const int waveRow = (wave >> 1) * 32;   // 4 wave-rows of 32
    const int waveCol = (wave & 1) * 64;    // 2 wave-cols of 64

    v8f acc[2][4];
#pragma unroll
    for (int m = 0; m < 2; ++m)
#pragma unroll
        for (int n = 0; n < 4; ++n)
            acc[m][n] = (v8f)0.0f;

    for (int kt = 0; kt < NMAT; kt += 32) {
        // ---- stage A: 128 rows x 32 k, float4 per thread x4 --------------
#pragma unroll
        for (int it = 0; it < 4; ++it) {
            int idx  = tid + it * 256;          // 0..1023 float4 slots
            int arow = idx >> 3;                // 8 float4 per row
            int ac4  = (idx & 7) << 2;
            const float4 v =
                *(const float4*)&Amat[(rowBase + arow) * NMAT + kt + ac4];
            *(float4*)&As[arow][ac4] = v;
        }
        // ---- stage B transposed: 32 k x 128 cols -> Bs[col][k] -----------
#pragma unroll
        for (int it = 0; it < 4; ++it) {
            int idx  = tid + it * 256;
            int krow = idx >> 5;                // 32 float4 per k-row
            int bc4  = (idx & 31) << 2;
            const float4 v =
                *(const float4*)&Bmat[(kt + krow) * NMAT + colBase + bc4];
            Bs[bc4 + 0][krow] = v.x;
            Bs[bc4 + 1][krow] = v.y;
            Bs[bc4 + 2][krow] = v.z;
            Bs[bc4 + 3][krow] = v.w;
        }
        __syncthreads();

        // ---- compute: 8 k-steps of 4, 8 WMMAs each -----------------------
#pragma unroll
        for (int kk = 0; kk < 32; kk += 4) {
            const int ka = kk + 2 * half;       // K pair per lane-half
            v2f a[2], b[4];
#pragma unroll
            for (int m = 0; m < 2; ++m)
                a[m] = *(const v2f*)&As[waveRow + 16 * m + r][ka];
#pragma unroll
            for (int n = 0; n < 4; ++n)
                b[n] = *(const v2f*)&Bs[waveCol + 16 * n + r][ka];
#pragma unroll
            for (int m = 0; m < 2; ++m)
#pragma unroll
                for (int n = 0; n < 4; ++n)
                    acc[m][n] = __builtin_amdgcn_wmma_f32_16x16x4_f32(
                        /*neg_a=*/false, a[m],
                        /*neg_b=*/false, b[n],
                        /*c_mod=*/(short)0, acc[m][n],
                        /*reuse_a=*/false, /*reuse_b=*/false);
        }
        __syncthreads();
    }

    // ---- epilogue: D = alpha*acc (+ I on diagonal) -----------------------
#pragma unroll
    for (int m = 0; m < 2; ++m) {
#pragma unroll
        for (int n = 0; n < 4; ++n) {
#pragma unroll
            for (int i = 0; i < 8; ++i) {
                const int row = rowBase + waveRow + 16 * m + i + 8 * half;
                const int col = colBase + waveCol + 16 * n + r;
                float val = alpha * acc[m][n][i];
                if (addIdentity && (row == col)) val += 1.0f;
                Dmat[row * NMAT + col] = val;
            }
        }
    }
}

// ---------------------------------------------------------------------------
extern "C" void kernel_launch(void* const* d_in, const int* in_sizes, int n_in,
                              void* d_out, int out_size, void* d_ws, size_t ws_size,
                              hipStream_t stream)
{
    (void)in_sizes; (void)n_in; (void)out_size; (void)ws_size;

    const float* A  = (const float*)d_in[0];
    float*       out = (float*)d_out;

    const size_t NN = (size_t)NMAT * NMAT;       // 4M floats = 16MB
    float* X  = (float*)d_ws;                     // ws needs 3 * 16MB = 48MB
    float* P1 = X + NN;
    float* P2 = P1 + NN;

    // X = (A - A^T)/16 ; P1 = I + X/8
    skew_init_kernel<<<(int)(NN / 256), 256, 0, stream>>>(A, X, P1);

    dim3 grid(NMAT / 128, NMAT / 128);   // 16 x 16 workgroups
    dim3 block(256);

    // Horner: P <- I + (1/k) * X * P, k = 7..1
    float* cur = P1;
    float* oth = P2;
    for (int k = HORNER_TOP - 1; k >= 1; --k) {
        gemm_wmma_f32_kernel<<<grid, block, 0, stream>>>(
            X, cur, oth, 1.0f / (float)k, /*addIdentity=*/1);
        float* t = cur; cur = oth; oth = t;
    }

    // Squarings: P <- P*P, last one lands in d_out
    for (int sq = 0; sq < SQUARINGS; ++sq) {
        float* dst = (sq == SQUARINGS - 1) ? out : oth;
        gemm_wmma_f32_kernel<<<grid, block, 0, stream>>>(
            cur, cur, dst, 1.0f, /*addIdentity=*/0);
        oth = cur;
        cur = dst;
    }
}